// RGCNSkipConnection_88974542504394
// MI455X (gfx1250) — compile-verified
//
#include <hip/hip_runtime.h>
#include <hip/hip_bf16.h>

#define N_NODES  50000
#define N_EDGES  600000
#define N_REL    6
#define D        128
#define N_GRAPHS 512
#define N_FEATX  13
#define G_FEAT   200
#define H1       512
#define H2       256
#define KREL     (N_REL * D)   // 768
#define KTOT     (KREL + D)    // 896
#define AP       900           // LDS row stride: 900%64=4 -> conflict-free, 16B aligned

typedef __attribute__((ext_vector_type(2))) float v2f;
typedef __attribute__((ext_vector_type(8))) float v8f;
typedef int v4i __attribute__((vector_size(16)));   // matches builtin param type

#define HAS_ASYNC_LDS __has_builtin(__builtin_amdgcn_global_load_async_to_lds_b128)

__device__ __forceinline__ void wait_async_zero() {
#if __has_builtin(__builtin_amdgcn_s_wait_asynccnt)
    __builtin_amdgcn_s_wait_asynccnt(0);
#else
    asm volatile("s_wait_asynccnt 0x0" ::: "memory");
#endif
}

// ---------------------------------------------------------------- zero (vec4)
__global__ void k_zero(float4* __restrict__ p, long n4) {
    long i = (long)blockIdx.x * blockDim.x + threadIdx.x;
    long s = (long)gridDim.x * blockDim.x;
    float4 z = make_float4(0.f, 0.f, 0.f, 0.f);
    for (; i < n4; i += s) p[i] = z;
}

// ---------------------------------------------------------------- encoder h = x @ encW + b
__global__ void __launch_bounds__(D) k_encode(const float* __restrict__ x,
                                              const float* __restrict__ W,
                                              const float* __restrict__ b,
                                              float* __restrict__ h) {
    __shared__ float sx[N_FEATX];
    int n = blockIdx.x;
    int d = threadIdx.x;
    if (d < N_FEATX) sx[d] = x[n * N_FEATX + d];
    __syncthreads();
    float acc = b[d];
#pragma unroll
    for (int k = 0; k < N_FEATX; ++k) acc += sx[k] * W[k * D + d];
    h[(size_t)n * D + d] = acc;
}

// ---------------------------------------------------------------- edge scatter: sums[(dst,r)] += h[src], cnt++
__global__ void __launch_bounds__(256) k_scatter(const int* __restrict__ ei,
                                                 const int* __restrict__ et,
                                                 const float* __restrict__ h,
                                                 float* __restrict__ sums,
                                                 float* __restrict__ cnt) {
    int e = blockIdx.x * 8 + (threadIdx.x >> 5);
    if (e >= N_EDGES) return;
    int lane = threadIdx.x & 31;
    int src = ei[e];
    int dst = ei[N_EDGES + e];
    int r   = et[e];
    const float4* hv = (const float4*)(h + (size_t)src * D);
    float4 v = hv[lane];
    float* sb = sums + ((size_t)dst * N_REL + r) * D + lane * 4;
    atomicAdd(sb + 0, v.x);
    atomicAdd(sb + 1, v.y);
    atomicAdd(sb + 2, v.z);
    atomicAdd(sb + 3, v.w);
    if (lane == 0) atomicAdd(cnt + (size_t)dst * N_REL + r, 1.0f);
}

// ---------------------------------------------------------------- RGCN layer: WMMA GEMM + bias + skip + PReLU + L2norm
// Per block: 16-node tile, 8 waves. A = [means(768) | h(128)] in LDS; B = [relW;rootW] from L2.
__global__ void __launch_bounds__(256)
k_rgcn(const float* __restrict__ sums, const float* __restrict__ cnt,
       const float* __restrict__ hin,
       const float* __restrict__ relW,   // [768, 128] this layer
       const float* __restrict__ rootW,  // [128, 128] this layer
       const float* __restrict__ bias,   // [128]
       const float* __restrict__ prelu_a,
       float* __restrict__ hout) {
    __shared__ __align__(16) float smem[16 * AP + 96 + 16];
    float* sA   = smem;                 // 16 x 896 A panel (stride AP)
    float* sCnt = smem + 16 * AP;       // 96 reciprocal counts
    float* sOut = smem;                 // reused after WMMA: 16 x 128 tile
    float* sNrm = smem + 16 * AP + 96;  // 16 row norms

    const int tid   = threadIdx.x;
    const int node0 = blockIdx.x * 16;

    // ---- skip/h columns (768..895): straight copy -> async DMA global->LDS if available
#if HAS_ASYNC_LDS
    for (int t = tid; t < 16 * D / 4; t += 256) {        // 512 x b128, 2 per thread
        int m = t >> 5;                                   // row
        int q = (t & 31) << 2;                            // float offset in row
        const float* g = hin + (size_t)(node0 + m) * D + q;
        float*       l = &sA[m * AP + KREL + q];
        __builtin_amdgcn_global_load_async_to_lds_b128((v4i*)g, (v4i*)l, 0, 0);
    }
#else
    for (int t = tid; t < 16 * D / 4; t += 256) {
        int m = t >> 5;
        int q = (t & 31) << 2;
        *(float4*)&sA[m * AP + KREL + q] =
            *(const float4*)(hin + (size_t)(node0 + m) * D + q);
    }
#endif

    if (tid < 96) {
        float c = cnt[(size_t)(node0 + tid / N_REL) * N_REL + (tid % N_REL)];
        sCnt[tid] = 1.0f / fmaxf(c, 1.0f);
    }
    __syncthreads();

    // ---- mean columns (0..767): vectorized load + scale + store
    for (int t = tid; t < 16 * KREL / 4; t += 256) {      // 3072 x float4, 12 per thread
        int m  = t / (KREL / 4);
        int q  = t - m * (KREL / 4);
        int kk = q << 2;
        float4 v = *(const float4*)(sums + (size_t)(node0 + m) * KREL + kk);
        float sc = sCnt[m * N_REL + (kk >> 7)];
        v.x *= sc; v.y *= sc; v.z *= sc; v.w *= sc;
        *(float4*)&sA[m * AP + kk] = v;
    }
#if HAS_ASYNC_LDS
    wait_async_zero();
#endif
    __syncthreads();

    const int wave = tid >> 5;
    const int lane = tid & 31;
    const int arow = lane & 15;
    const int koff = (lane >> 4) << 1;          // 0 for lanes 0-15, 2 for 16-31
    const int col  = (wave << 4) + (lane & 15); // output column of this lane

    v8f acc = {};
    // relation-weight portion (K = 0..767)
    for (int kk = 0; kk < KREL; kk += 4) {
        int k0 = kk + koff;
        v2f a, b;
        a.x = sA[arow * AP + k0];
        a.y = sA[arow * AP + k0 + 1];
        const float* wp = relW + (size_t)k0 * D + col;
        b.x = wp[0];
        b.y = wp[D];
        acc = __builtin_amdgcn_wmma_f32_16x16x4_f32(false, a, false, b,
                                                    (short)0, acc, false, false);
    }
    // root-weight portion (K = 768..895)
    for (int kk = KREL; kk < KTOT; kk += 4) {
        int k0 = kk + koff;
        v2f a, b;
        a.x = sA[arow * AP + k0];
        a.y = sA[arow * AP + k0 + 1];
        const float* wp = rootW + (size_t)(k0 - KREL) * D + col;
        b.x = wp[0];
        b.y = wp[D];
        acc = __builtin_amdgcn_wmma_f32_16x16x4_f32(false, a, false, b,
                                                    (short)0, acc, false, false);
    }
    __syncthreads();  // all waves done with WMMA reads of sA

    // ---- grab skip values from LDS (columns 768..895 of A panel) before sOut overwrites
    const int rbase = (lane >> 4) << 3;    // C layout: VGPR j -> row j (lanes<16) / j+8
    float skipv[8];
#pragma unroll
    for (int j = 0; j < 8; ++j)
        skipv[j] = sA[(rbase + j) * AP + KREL + col];
    __syncthreads();  // everyone captured skip; safe to overwrite smem

    // ---- epilogue: + bias + skip, PReLU; stage to LDS for row L2 norms
    const float alpha = prelu_a[0];
    const float bb    = bias[col];
#pragma unroll
    for (int j = 0; j < 8; ++j) {
        int row = rbase + j;
        float v = acc[j] + bb + skipv[j];
        v = (v >= 0.f) ? v : alpha * v;
        sOut[row * D + col] = v;
    }
    __syncthreads();

    if (tid < 16) {
        float s = 0.f;
        const float4* rp = (const float4*)&sOut[tid * D];
#pragma unroll 8
        for (int q = 0; q < D / 4; ++q) {
            float4 v = rp[q];
            s += v.x * v.x + v.y * v.y + v.z * v.z + v.w * v.w;
        }
        sNrm[tid] = 1.0f / fmaxf(sqrtf(s), 1e-12f);
    }
    __syncthreads();

    float4* op = (float4*)(hout + (size_t)node0 * D);
    const float4* sp = (const float4*)sOut;
    for (int q = tid; q < 16 * D / 4; q += 256) {
        float4 v = sp[q];
        float n = sNrm[q >> 5];
        v.x *= n; v.y *= n; v.z *= n; v.w *= n;
        op[q] = v;
    }
}

// ---------------------------------------------------------------- mean pool per graph
__global__ void __launch_bounds__(256) k_pool(const float* __restrict__ h,
                                              const int* __restrict__ batch,
                                              float* __restrict__ gsum,
                                              float* __restrict__ gcnt) {
    size_t i = (size_t)blockIdx.x * blockDim.x + threadIdx.x;
    if (i >= (size_t)N_NODES * D) return;
    int n = (int)(i >> 7);
    int d = (int)(i & 127);
    int g = batch[n];
    atomicAdd(&gsum[(size_t)g * D + d], h[i]);
    if (d == 0) atomicAdd(&gcnt[g], 1.0f);
}

// ---------------------------------------------------------------- MLP head
__global__ void __launch_bounds__(H1) k_mlp1(const float* __restrict__ gsum,
                                             const float* __restrict__ gcnt,
                                             const float* __restrict__ feat,
                                             const float* __restrict__ W1,
                                             const float* __restrict__ b1,
                                             float* __restrict__ z1) {
    __shared__ float zr[D + G_FEAT];
    int i = blockIdx.x, j = threadIdx.x;
    if (j < D) zr[j] = gsum[(size_t)i * D + j] / fmaxf(gcnt[i], 1.0f);
    else if (j < D + G_FEAT) zr[j] = feat[(size_t)i * G_FEAT + (j - D)];
    __syncthreads();
    float a = b1[j];
    for (int k = 0; k < D + G_FEAT; ++k) a += zr[k] * W1[k * H1 + j];
    z1[(size_t)i * H1 + j] = fmaxf(a, 0.f);
}

__global__ void __launch_bounds__(H2) k_mlp2(const float* __restrict__ z1,
                                             const float* __restrict__ W2,
                                             const float* __restrict__ b2,
                                             float* __restrict__ z2) {
    __shared__ float zr[H1];
    int i = blockIdx.x, j = threadIdx.x;
    zr[j]       = z1[(size_t)i * H1 + j];
    zr[j + 256] = z1[(size_t)i * H1 + j + 256];
    __syncthreads();
    float a = b2[j];
    for (int k = 0; k < H1; ++k) a += zr[k] * W2[k * H2 + j];
    z2[(size_t)i * H2 + j] = fmaxf(a, 0.f);
}

__global__ void k_out(const float* __restrict__ z2, const float* __restrict__ oW,
                      const float* __restrict__ ob, float* __restrict__ out) {
    int i = blockIdx.x * blockDim.x + threadIdx.x;
    if (i >= N_GRAPHS) return;
    float a = ob[0];
    for (int k = 0; k < H2; ++k) a += z2[(size_t)i * H2 + k] * oW[k];
    out[i] = a;
}

// ---------------------------------------------------------------- launch
extern "C" void kernel_launch(void* const* d_in, const int* in_sizes, int n_in,
                              void* d_out, int out_size, void* d_ws, size_t ws_size,
                              hipStream_t stream) {
    (void)in_sizes; (void)n_in; (void)out_size; (void)ws_size;
    const float* x     = (const float*)d_in[0];
    const int*   ei    = (const int*)d_in[1];
    const int*   et    = (const int*)d_in[2];
    const float* feat  = (const float*)d_in[3];
    const int*   batch = (const int*)d_in[4];
    const float* encW  = (const float*)d_in[5];
    const float* encB  = (const float*)d_in[6];
    const float* relW  = (const float*)d_in[7];
    const float* rootW = (const float*)d_in[8];
    const float* convB = (const float*)d_in[9];
    const float* pa    = (const float*)d_in[10];
    const float* W1    = (const float*)d_in[11];
    const float* b1    = (const float*)d_in[12];
    const float* W2    = (const float*)d_in[13];
    const float* b2    = (const float*)d_in[14];
    const float* oW    = (const float*)d_in[15];
    const float* ob    = (const float*)d_in[16];
    float* out = (float*)d_out;

    char* ws = (char*)d_ws;
    size_t off = 0;
    auto alloc = [&](size_t bytes) -> void* {
        void* p = ws + off;
        off = (off + bytes + 255) & ~(size_t)255;
        return p;
    };
    float* h0   = (float*)alloc((size_t)N_NODES * D * 4);
    float* h1   = (float*)alloc((size_t)N_NODES * D * 4);
    float* sums = (float*)alloc((size_t)N_NODES * N_REL * D * 4);  // cnt follows contiguously
    float* cntb = (float*)alloc((size_t)N_NODES * N_REL * 4);
    float* gsum = (float*)alloc((size_t)N_GRAPHS * D * 4);         // gcnt follows contiguously
    float* gcnt = (float*)alloc((size_t)N_GRAPHS * 4);
    float* z1   = (float*)alloc((size_t)N_GRAPHS * H1 * 4);
    float* z2   = (float*)alloc((size_t)N_GRAPHS * H2 * 4);

    k_encode<<<N_NODES, D, 0, stream>>>(x, encW, encB, h0);

    float* hin = h0;
    float* hout = h1;
    for (int l = 0; l < 3; ++l) {
        long nz4 = (long)N_NODES * N_REL * (D + 1) / 4;  // sums + cnt, contiguous
        k_zero<<<(unsigned)((nz4 + 255) / 256), 256, 0, stream>>>((float4*)sums, nz4);
        k_scatter<<<(N_EDGES + 7) / 8, 256, 0, stream>>>(ei, et, hin, sums, cntb);
        k_rgcn<<<N_NODES / 16, 256, 0, stream>>>(
            sums, cntb, hin,
            relW + (size_t)l * KREL * D,
            rootW + (size_t)l * D * D,
            convB + (size_t)l * D, pa, hout);
        float* t = hin; hin = hout; hout = t;
    }
    // hin now holds the final node embeddings

    long ng4 = (long)(N_GRAPHS * D + N_GRAPHS) / 4;  // gsum + gcnt, contiguous
    k_zero<<<(unsigned)((ng4 + 255) / 256), 256, 0, stream>>>((float4*)gsum, ng4);
    k_pool<<<(N_NODES * D) / 256, 256, 0, stream>>>(hin, batch, gsum, gcnt);
    k_mlp1<<<N_GRAPHS, H1, 0, stream>>>(gsum, gcnt, feat, W1, b1, z1);
    k_mlp2<<<N_GRAPHS, H2, 0, stream>>>(z1, W2, b2, z2);
    k_out<<<(N_GRAPHS + 255) / 256, 256, 0, stream>>>(z2, oW, ob, out);
}